// STKEncoder_1322849927633
// MI455X (gfx1250) — compile-verified
//
#include <hip/hip_runtime.h>
#include <float.h>
#include <math.h>

// ---------------- problem dimensions ----------------
#define BSZ  8
#define KC   8
#define NCV  16
#define TT   3072
#define WGT  64
#define MM   (BSZ*KC)      // 64
#define HH   256
#define WCC  256
#define RHH  1024
#define RWW  512

#define H1   1024
#define W1   31
#define H2   341
#define W2C  15
#define H3   113
#define W3C  7

#define PBLK 2048          // partial blocks per channel for stats

// ---------------- workspace layout (float offsets) ----------------
#define OFF_PA    0                       // 2ch * 2048 partial sums
#define OFF_PB    4096                    // 2ch * 2048 partial sumsq
#define OFF_SCAL  8192                    // scalars
#define OFF_Y1    8256
#define N_Y1      (MM*H1*W1)              // 2,031,616
#define OFF_MP3   (OFF_Y1 + N_Y1)
#define OFF_MP5   (OFF_MP3 + N_Y1)
#define OFF_Y2    (OFF_MP5 + N_Y1)
#define N_Y2      (MM*2*H2*W2C)           // 654,720
#define OFF_Y3    (OFF_Y2 + N_Y2)
#define N_Y3      (MM*H3*W3C)             // 50,624
#define OFF_GENR  (OFF_Y3 + N_Y3)
#define N_GENR    (MM*RHH)                // 65,536
#define OFF_MLO   (OFF_GENR + N_GENR)
#define N_MLO     (MM*HH*WCC)             // 4,194,304
#define OFF_MIX   OFF_Y1                  // aliases dead y1/mp3 region
#define N_MIX     (BSZ*RHH*RWW)           // 4,194,304
#define OFF_YF    (OFF_MIX + N_MIX)       // aliases dead mp5/y2/y3/genr/mlo

// scalar slots
#define S_GMAX 0
#define S_M1   2
#define S_V1   3
#define S_M2   4   // 2 entries
#define S_V2   6   // 2 entries
#define S_M3   8
#define S_V3   9
#define S_MF   10
#define S_VF   11

typedef __attribute__((ext_vector_type(2))) float v2f;
typedef __attribute__((ext_vector_type(8))) float v8f;

// ---------------- reductions (blockDim.x == 256) ----------------
__device__ __forceinline__ void block_reduce2(float a, float b, float& ra, float& rb) {
    __shared__ float sa[256], sb[256];
    sa[threadIdx.x] = a; sb[threadIdx.x] = b; __syncthreads();
    for (int off = 128; off >= 1; off >>= 1) {
        if ((int)threadIdx.x < off) {
            sa[threadIdx.x] += sa[threadIdx.x + off];
            sb[threadIdx.x] += sb[threadIdx.x + off];
        }
        __syncthreads();
    }
    ra = sa[0]; rb = sb[0];
}

__device__ __forceinline__ float block_reduce_max(float v) {
    __shared__ float sm[256];
    sm[threadIdx.x] = v; __syncthreads();
    for (int off = 128; off >= 1; off >>= 1) {
        if ((int)threadIdx.x < off)
            sm[threadIdx.x] = fmaxf(sm[threadIdx.x], sm[threadIdx.x + off]);
        __syncthreads();
    }
    return sm[0];
}

__device__ __forceinline__ float lrelu(float x) { return x >= 0.f ? x : 0.1f * x; }

// ---------------- 1) global max|gather| ----------------
__global__ void k_absmax_part(const float* __restrict__ g, int n, float* part) {
    float m = 0.f;
    for (int i = blockIdx.x * 256 + threadIdx.x; i < n; i += gridDim.x * 256)
        m = fmaxf(m, fabsf(g[i]));
    float r = block_reduce_max(m);
    if (threadIdx.x == 0) part[blockIdx.x] = r;
}

__global__ void k_absmax_final(const float* part, int np, float* scal) {
    float m = 0.f;
    for (int i = threadIdx.x; i < np; i += 256) m = fmaxf(m, part[i]);
    float r = block_reduce_max(m);
    if (threadIdx.x == 0) scal[S_GMAX] = r;
}

// ---------------- 2) CV mask (64,256,256) via jnp.interp ----------------
__global__ void k_cvmask(const float* __restrict__ cv, const float* __restrict__ vmm,
                         float* __restrict__ mlo) {
    int m = blockIdx.x;          // curve 0..63
    int h = threadIdx.x;         // time row 0..255
    int bs = m >> 3;
    float vmin = vmm[bs * 2 + 0], vmax = vmm[bs * 2 + 1];
    float dt = 7000.f / 255.f;
    float dv = (vmax - vmin) / 255.f;
    const float* c0 = cv + m * NCV * 2;
    float tq = (float)h;
    float x0 = c0[0] / dt, f0 = (c0[1] - vmin) / dv;
    float xn = c0[(NCV - 1) * 2] / dt, fn = (c0[(NCV - 1) * 2 + 1] - vmin) / dv;
    float v;
    if (tq <= x0)      v = f0;
    else if (tq >= xn) v = fn;
    else {
        v = fn;
        float xp = x0, fp = f0;
        for (int n = 1; n < NCV; ++n) {
            float xc = c0[n * 2] / dt, fc = (c0[n * 2 + 1] - vmin) / dv;
            if (tq <= xc) { v = fp + (tq - xp) * (fc - fp) / (xc - xp); break; }
            xp = xc; fp = fc;
        }
    }
    int vi = (int)v; vi = min(max(vi, 0), WCC - 1);
    float* row = mlo + ((size_t)m * HH + h) * WCC;
    for (int c = 0; c < WCC; ++c) row[c] = 0.01f + (c == vi ? 0.9f : 0.f);
}

// ---------------- 3) conv1 (1->1ch, 3x3, stride 3x2) + stat partials ----------------
__global__ void k_conv1(const float* __restrict__ g, const float* __restrict__ scal,
                        const float* __restrict__ w1, const float* __restrict__ b1,
                        float* __restrict__ y1, float* pa, float* pb) {
    float inv = 1.f / scal[S_GMAX];
    float w[9];
#pragma unroll
    for (int t = 0; t < 9; ++t) w[t] = w1[t];
    float bb = b1[0], s = 0.f, s2 = 0.f;
    const int total = N_Y1;
    for (int idx = blockIdx.x * 256 + threadIdx.x; idx < total; idx += gridDim.x * 256) {
        int j = idx % W1, rest = idx / W1;
        int i = rest % H1, m = rest / H1;
        const float* base = g + (size_t)m * (TT * WGT) + (3 * i) * WGT + 2 * j;
        float acc = bb;
#pragma unroll
        for (int r = 0; r < 3; ++r)
#pragma unroll
            for (int c = 0; c < 3; ++c)
                acc += w[r * 3 + c] * (base[r * WGT + c] * inv);
        y1[idx] = acc;
        s += acc; s2 += acc * acc;
    }
    float ra, rb; block_reduce2(s, s2, ra, rb);
    if (threadIdx.x == 0) { pa[blockIdx.x] = ra; pb[blockIdx.x] = rb; }
}

// ---------------- stats finalize (one block per channel) ----------------
__global__ void k_stats_final(const float* pa, const float* pb, int nblk, float n,
                              float* scal, int mi, int vi) {
    int ch = blockIdx.x;
    float s = 0.f, s2 = 0.f;
    for (int t = threadIdx.x; t < nblk; t += 256) { s += pa[ch * nblk + t]; s2 += pb[ch * nblk + t]; }
    float ra, rb; block_reduce2(s, s2, ra, rb);
    if (threadIdx.x == 0) {
        float mean = ra / n;
        float var = rb / n - mean * mean;
        scal[mi + ch] = mean;
        scal[vi + ch] = fmaxf(var, 0.f);
    }
}

// ---------------- BN (train-mode) + LeakyReLU, in place ----------------
__global__ void k_bn_lrelu(float* __restrict__ y, int total, int hw, int nch,
                           const float* __restrict__ gamma, const float* __restrict__ beta,
                           const float* __restrict__ scal, int mi, int vi) {
    int idx = blockIdx.x * 256 + threadIdx.x;
    if (idx >= total) return;
    int ch = (idx / hw) % nch;
    float m = scal[mi + ch], v = scal[vi + ch];
    float x = (y[idx] - m) * (1.f / sqrtf(v + 1e-5f)) * gamma[ch] + beta[ch];
    y[idx] = lrelu(x);
}

// ---------------- maxpool 3 and 5 (stride 1, -inf pad) ----------------
__global__ void k_maxpool(const float* __restrict__ xa, float* __restrict__ mp3,
                          float* __restrict__ mp5) {
    int idx = blockIdx.x * 256 + threadIdx.x;
    if (idx >= N_Y1) return;
    int j = idx % W1, rest = idx / W1;
    int i = rest % H1, m = rest / H1;
    float m3 = -FLT_MAX, m5 = -FLT_MAX;
    for (int di = -2; di <= 2; ++di) {
        int ii = i + di; if (ii < 0 || ii >= H1) continue;
        for (int dj = -2; dj <= 2; ++dj) {
            int jj = j + dj; if (jj < 0 || jj >= W1) continue;
            float v = xa[((size_t)m * H1 + ii) * W1 + jj];
            m5 = fmaxf(m5, v);
            if (di >= -1 && di <= 1 && dj >= -1 && dj <= 1) m3 = fmaxf(m3, v);
        }
    }
    mp3[idx] = m3; mp5[idx] = m5;
}

// ---------------- conv2 (3->2ch, 3x3, stride 3x2) + per-ch stat partials ----------------
__global__ void k_conv2(const float* __restrict__ xa, const float* __restrict__ mp3,
                        const float* __restrict__ mp5, const float* __restrict__ w2,
                        const float* __restrict__ b2, float* __restrict__ y2,
                        float* pa, float* pb) {
    int oc = blockIdx.y;
    float w[27];
#pragma unroll
    for (int t = 0; t < 27; ++t) w[t] = w2[oc * 27 + t];
    float bb = b2[oc], s = 0.f, s2 = 0.f;
    const float* chans[3] = {xa, mp3, mp5};
    const int chtotal = MM * H2 * W2C;
    for (int idx = blockIdx.x * 256 + threadIdx.x; idx < chtotal; idx += gridDim.x * 256) {
        int j = idx % W2C, rest = idx / W2C;
        int i = rest % H2, m = rest / H2;
        float acc = bb;
#pragma unroll
        for (int ic = 0; ic < 3; ++ic) {
            const float* ch = chans[ic] + ((size_t)m * H1 + 3 * i) * W1 + 2 * j;
#pragma unroll
            for (int r = 0; r < 3; ++r)
#pragma unroll
                for (int c = 0; c < 3; ++c)
                    acc += w[(ic * 3 + r) * 3 + c] * ch[r * W1 + c];
        }
        y2[((size_t)(m * 2 + oc) * H2 + i) * W2C + j] = acc;
        s += acc; s2 += acc * acc;
    }
    float ra, rb; block_reduce2(s, s2, ra, rb);
    if (threadIdx.x == 0) { pa[oc * PBLK + blockIdx.x] = ra; pb[oc * PBLK + blockIdx.x] = rb; }
}

// ---------------- conv3 (2->1ch, 3x2, stride 3x2) ----------------
__global__ void k_conv3(const float* __restrict__ y2, const float* __restrict__ w3,
                        const float* __restrict__ b3, float* __restrict__ y3,
                        float* pa, float* pb) {
    float w[12];
#pragma unroll
    for (int t = 0; t < 12; ++t) w[t] = w3[t];
    float bb = b3[0], s = 0.f, s2 = 0.f;
    const int total = N_Y3;
    for (int idx = blockIdx.x * 256 + threadIdx.x; idx < total; idx += gridDim.x * 256) {
        int j = idx % W3C, rest = idx / W3C;
        int i = rest % H3, m = rest / H3;
        float acc = bb;
#pragma unroll
        for (int ic = 0; ic < 2; ++ic)
#pragma unroll
            for (int r = 0; r < 3; ++r)
#pragma unroll
                for (int c = 0; c < 2; ++c)
                    acc += w[(ic * 3 + r) * 2 + c] *
                           y2[((size_t)(m * 2 + ic) * H2 + 3 * i + r) * W2C + 2 * j + c];
        y3[idx] = acc;
        s += acc; s2 += acc * acc;
    }
    float ra, rb; block_reduce2(s, s2, ra, rb);
    if (threadIdx.x == 0) { pa[blockIdx.x] = ra; pb[blockIdx.x] = rb; }
}

// ---------------- genr: resize (113,7) -> (1024,1), half-pixel, antialias on W ----------------
__global__ void k_genr(const float* __restrict__ xe, float* __restrict__ genr) {
    int idx = blockIdx.x * 256 + threadIdx.x;
    if (idx >= N_GENR) return;
    int h = idx % RHH, m = idx / RHH;
    float cc = (h + 0.5f) * (113.f / 1024.f) - 0.5f;   // upsample: plain bilinear
    float fl = floorf(cc);
    float f = cc - fl;
    int r0 = min(max((int)fl, 0), H3 - 1);
    int r1 = min(max((int)fl + 1, 0), H3 - 1);
    float acc = 0.f;
#pragma unroll
    for (int j = 0; j < 7; ++j) {
        float wj = (7.f - fabsf((float)(j - 3))) / 37.f;  // antialiased triangle, 7->1
        acc += wj * ((1.f - f) * xe[((size_t)m * H3 + r0) * W3C + j] +
                     f         * xe[((size_t)m * H3 + r1) * W3C + j]);
    }
    genr[idx] = acc;
}

// ---------------- mix via V_WMMA_F32_16X16X4_F32 ----------------
// mix[b,h,w] = sum_k genr[b,k,h] * BilinearResize(mlo[b,k])(h,w)
// Per 16h x 16w tile: K-dim = (curve k, h-source row rr) pairs, 8*6 = 48
//   -> 12 fully unrolled chained WMMAs, all fragments preloaded into registers.
// h-bilinear weights folded into A; w-bilinear interpolation folded into B.
__global__ void k_mix_wmma(const float* __restrict__ genr, const float* __restrict__ mlo,
                           float* __restrict__ mix) {
    const int lane = threadIdx.x;          // 0..31, one wave per block, EXEC all ones
    const int w0 = blockIdx.x * 16;
    const int h0 = blockIdx.y * 16;
    const int b  = blockIdx.z;
    const int i  = lane & 15;
    const int hi = lane >> 4;              // 0 = lanes 0-15, 1 = lanes 16-31
    const int R0 = (h0 >> 2) - 1;          // lowest candidate source row

    // A-row (M = i) h-interp weights: c = (h+0.5)*(256/1024) - 0.5
    const int h = h0 + i;
    const float c = h * 0.25f - 0.375f;
    const float fl = floorf(c);
    const float f  = c - fl;
    const int r_lo = min(max((int)fl, 0), HH - 1);
    const int r_hi = min(max((int)fl + 1, 0), HH - 1);

    // B-column (N = i) w-interp: c = (w+0.5)*(256/512) - 0.5
    const int wcol = w0 + i;
    const float cw = wcol * 0.5f - 0.25f;
    const float wl = floorf(cw);
    const float fw = cw - wl;
    const int c_lo = min(max((int)wl, 0), WCC - 1);
    const int c_hi = min(max((int)wl + 1, 0), WCC - 1);

    // hoist: genr values for all 8 curves at this lane's row h
    float gk[KC];
#pragma unroll
    for (int k = 0; k < KC; ++k)
        gk[k] = genr[(size_t)(b * KC + k) * RHH + h];

    // hoist: h-bilinear weight for each candidate source-row offset rr
    float wArr[6];
#pragma unroll
    for (int rr = 0; rr < 6; ++rr) {
        const int r = R0 + rr;
        float wA = 0.f;
        if (r_lo == r) wA += (1.f - f);
        if (r_hi == r) wA += f;
        wArr[rr] = wA;
    }

    // precompute this lane's 24 A and 24 B fragment elements (one load clause),
    // then issue the 12 WMMAs back-to-back on registers only.
    float Av[24], Bv[24];
#pragma unroll
    for (int g = 0; g < 12; ++g) {
#pragma unroll
        for (int v = 0; v < 2; ++v) {
            // lane<16 holds local K = {0,1}; lane>=16 holds local K = {2,3}
            const int idx = 4 * g + 2 * hi + v;    // global K index 0..47
            const int rr = idx >> 3;               // source-row offset 0..5
            const int k  = idx & 7;                // curve
            Av[2 * g + v] = gk[k] * wArr[rr];
            const int src = min(max(R0 + rr, 0), HH - 1);
            const float* row = mlo + ((size_t)(b * KC + k) * HH + src) * WCC;
            Bv[2 * g + v] = (1.f - fw) * row[c_lo] + fw * row[c_hi];
        }
    }

    v8f acc = {};
#pragma unroll
    for (int g = 0; g < 12; ++g) {
        v2f A, B;
        A[0] = Av[2 * g]; A[1] = Av[2 * g + 1];
        B[0] = Bv[2 * g]; B[1] = Bv[2 * g + 1];
        acc = __builtin_amdgcn_wmma_f32_16x16x4_f32(
            /*neg_a=*/false, A, /*neg_b=*/false, B,
            /*c_mod=*/(short)0, acc, /*reuse_a=*/false, /*reuse_b=*/false);
    }

    // C/D layout: VGPR g -> (M=g, N=lane) for lanes 0-15, (M=g+8, N=lane-16) for 16-31
#pragma unroll
    for (int g = 0; g < 8; ++g) {
        const int row = h0 + g + hi * 8;
        mix[((size_t)b * RHH + row) * RWW + w0 + i] = acc[g];
    }
}

// ---------------- final conv 3x3 pad1 + stat partials ----------------
__global__ void k_convf(const float* __restrict__ mixv, const float* __restrict__ wf,
                        const float* __restrict__ bf, float* __restrict__ yf,
                        float* pa, float* pb) {
    float w[9];
#pragma unroll
    for (int t = 0; t < 9; ++t) w[t] = wf[t];
    float bb = bf[0], s = 0.f, s2 = 0.f;
    const int total = N_MIX;
    for (int idx = blockIdx.x * 256 + threadIdx.x; idx < total; idx += gridDim.x * 256) {
        int j = idx % RWW, rest = idx / RWW;
        int i = rest % RHH, bq = rest / RHH;
        float acc = bb;
#pragma unroll
        for (int r = 0; r < 3; ++r) {
            int ii = i - 1 + r; if (ii < 0 || ii >= RHH) continue;
#pragma unroll
            for (int c = 0; c < 3; ++c) {
                int jj = j - 1 + c; if (jj < 0 || jj >= RWW) continue;
                acc += w[r * 3 + c] * mixv[((size_t)bq * RHH + ii) * RWW + jj];
            }
        }
        yf[idx] = acc;
        s += acc; s2 += acc * acc;
    }
    float ra, rb; block_reduce2(s, s2, ra, rb);
    if (threadIdx.x == 0) { pa[blockIdx.x] = ra; pb[blockIdx.x] = rb; }
}

// ---------------- final BN + LeakyReLU -> d_out ----------------
__global__ void k_out(const float* __restrict__ yf, const float* __restrict__ gf,
                      const float* __restrict__ bef, const float* __restrict__ scal,
                      float* __restrict__ out) {
    int idx = blockIdx.x * 256 + threadIdx.x;
    if (idx >= N_MIX) return;
    float m = scal[S_MF], v = scal[S_VF];
    float x = (yf[idx] - m) * (1.f / sqrtf(v + 1e-5f)) * gf[0] + bef[0];
    out[idx] = lrelu(x);
}

// ---------------- launch ----------------
extern "C" void kernel_launch(void* const* d_in, const int* in_sizes, int n_in,
                              void* d_out, int out_size, void* d_ws, size_t ws_size,
                              hipStream_t stream) {
    const float* gather = (const float*)d_in[0];
    const float* cv     = (const float*)d_in[1];
    const float* vmm    = (const float*)d_in[2];
    const float* w1 = (const float*)d_in[3];  const float* b1 = (const float*)d_in[4];
    const float* g1 = (const float*)d_in[5];  const float* be1 = (const float*)d_in[6];
    const float* w2 = (const float*)d_in[7];  const float* b2 = (const float*)d_in[8];
    const float* g2 = (const float*)d_in[9];  const float* be2 = (const float*)d_in[10];
    const float* w3 = (const float*)d_in[11]; const float* b3 = (const float*)d_in[12];
    const float* g3 = (const float*)d_in[13]; const float* be3 = (const float*)d_in[14];
    const float* wf = (const float*)d_in[15]; const float* bf = (const float*)d_in[16];
    const float* gf = (const float*)d_in[17]; const float* bef = (const float*)d_in[18];
    float* out = (float*)d_out;
    float* ws = (float*)d_ws;

    float* pa   = ws + OFF_PA;
    float* pb   = ws + OFF_PB;
    float* scal = ws + OFF_SCAL;
    float* y1   = ws + OFF_Y1;
    float* mp3  = ws + OFF_MP3;
    float* mp5  = ws + OFF_MP5;
    float* y2   = ws + OFF_Y2;
    float* y3   = ws + OFF_Y3;
    float* genr = ws + OFF_GENR;
    float* mlo  = ws + OFF_MLO;
    float* mixv = ws + OFF_MIX;
    float* yf   = ws + OFF_YF;

    const int NG = BSZ * KC * TT * WGT;

    // 1) global max|gather|
    k_absmax_part<<<PBLK, 256, 0, stream>>>(gather, NG, pa);
    k_absmax_final<<<1, 256, 0, stream>>>(pa, PBLK, scal);

    // 2) CV mask (runs independently)
    k_cvmask<<<MM, 256, 0, stream>>>(cv, vmm, mlo);

    // 3) conv1 + BN + lrelu
    k_conv1<<<PBLK, 256, 0, stream>>>(gather, scal, w1, b1, y1, pa, pb);
    k_stats_final<<<1, 256, 0, stream>>>(pa, pb, PBLK, (float)N_Y1, scal, S_M1, S_V1);
    k_bn_lrelu<<<(N_Y1 + 255) / 256, 256, 0, stream>>>(y1, N_Y1, H1 * W1, 1, g1, be1, scal, S_M1, S_V1);

    // 4) maxpools + conv2 + BN + lrelu
    k_maxpool<<<(N_Y1 + 255) / 256, 256, 0, stream>>>(y1, mp3, mp5);
    k_conv2<<<dim3(PBLK, 2), 256, 0, stream>>>(y1, mp3, mp5, w2, b2, y2, pa, pb);
    k_stats_final<<<2, 256, 0, stream>>>(pa, pb, PBLK, (float)(MM * H2 * W2C), scal, S_M2, S_V2);
    k_bn_lrelu<<<(N_Y2 + 255) / 256, 256, 0, stream>>>(y2, N_Y2, H2 * W2C, 2, g2, be2, scal, S_M2, S_V2);

    // 5) conv3 + BN + lrelu
    k_conv3<<<PBLK, 256, 0, stream>>>(y2, w3, b3, y3, pa, pb);
    k_stats_final<<<1, 256, 0, stream>>>(pa, pb, PBLK, (float)N_Y3, scal, S_M3, S_V3);
    k_bn_lrelu<<<(N_Y3 + 255) / 256, 256, 0, stream>>>(y3, N_Y3, H3 * W3C, 1, g3, be3, scal, S_M3, S_V3);

    // 6) genr resize
    k_genr<<<(N_GENR + 255) / 256, 256, 0, stream>>>(y3, genr);

    // 7) WMMA mix: sum_k genr * resized mask  (one wave per 16x16 tile)
    k_mix_wmma<<<dim3(RWW / 16, RHH / 16, BSZ), 32, 0, stream>>>(genr, mlo, mixv);

    // 8) final conv + BN + lrelu
    k_convf<<<PBLK, 256, 0, stream>>>(mixv, wf, bf, yf, pa, pb);
    k_stats_final<<<1, 256, 0, stream>>>(pa, pb, PBLK, (float)N_MIX, scal, S_MF, S_VF);
    k_out<<<(N_MIX + 255) / 256, 256, 0, stream>>>(yf, gf, bef, scal, out);
}